// GeoGNNLayer_38104949850638
// MI455X (gfx1250) — compile-verified
//
#include <hip/hip_runtime.h>
#include <hip/hip_bf16.h>

// ---------------------------------------------------------------------------
// GeoGNN layer for MI455X (gfx1250, wave32).
// fp32 end-to-end; MLP GEMMs on V_WMMA_F32_16X16X4_F32 (fp32-exact matrix pipe).
// ---------------------------------------------------------------------------

typedef __attribute__((ext_vector_type(2))) float v2f;
typedef __attribute__((ext_vector_type(8))) float v8f;

#define N_ATOMS      100000
#define N_BONDS_DIR  200000
#define N_BONDS      100000
#define N_ANGLES     400000
#define N_GRAPHS     2048
#define DD           128
#define HH           256
#define GAMMA        10.0f
#define LN_EPS       1e-5f

__device__ __forceinline__ v8f wmma_f32(v2f a, v2f b, v8f c) {
    // 8-arg form: (neg_a, A, neg_b, B, c_mod, C, reuse_a, reuse_b)
    return __builtin_amdgcn_wmma_f32_16x16x4_f32(
        false, a, false, b, (short)0, c, false, false);
}

// -------------------- per-graph node counts (for SqrtGraphNorm) ------------
__global__ __launch_bounds__(256) void count_kernel(
    const int* __restrict__ gid, float* __restrict__ counts, int n) {
    int i = blockIdx.x * blockDim.x + threadIdx.x;
    if (i < n) atomicAdd(&counts[gid[i]], 1.0f);
}

// -------------------- atom-bond message scatter ----------------------------
// msg = node_feats[src] + edge_feats; agg[dst] += msg   (one thread = 4 chans)
__global__ __launch_bounds__(256) void scatter_ab_kernel(
    const float* __restrict__ nf, const float* __restrict__ ef,
    const int* __restrict__ src, const int* __restrict__ dst,
    float* __restrict__ agg, int nE) {
    long long idx = (long long)blockIdx.x * blockDim.x + threadIdx.x;
    long long total = (long long)nE * 32;            // 128/4 chunks per edge
    if (idx >= total) return;
    int e  = (int)(idx >> 5);
    int c4 = (int)(idx & 31) * 4;
    int s = src[e], d = dst[e];
    const float4 a = *(const float4*)(nf + (size_t)s * DD + c4);
    const float4 b = *(const float4*)(ef + (size_t)e * DD + c4);
    float* p = agg + (size_t)d * DD + c4;
    atomicAdd(p + 0, a.x + b.x);
    atomicAdd(p + 1, a.y + b.y);
    atomicAdd(p + 2, a.z + b.z);
    atomicAdd(p + 3, a.w + b.w);
}

// -------------------- bond embedding ---------------------------------------
// bond_embed[j] = sum_i bond_emb[i][cat[2j,i]] + RBF(bond_float[2j]) @ W + b
__global__ __launch_bounds__(128) void bond_embed_kernel(
    const float* __restrict__ bond_emb,   // [3,16,128]
    const int*   __restrict__ bond_cat,   // [200000,3]
    const float* __restrict__ bond_float, // [200000]
    const float* __restrict__ rbfW,       // [20,128]
    const float* __restrict__ rbfb,       // [128]
    float* __restrict__ out, int nB) {
    __shared__ float e[20];
    int j = blockIdx.x;
    int d = threadIdx.x;
    int row = 2 * j;                      // [::2] on directed bonds
    if (d < 20) {
        float x = bond_float[row];
        float df = x - 0.1f * (float)d;   // centers 0.0 .. 1.9
        e[d] = expf(-GAMMA * df * df);
    }
    __syncthreads();
    int c0 = bond_cat[row * 3 + 0];
    int c1 = bond_cat[row * 3 + 1];
    int c2 = bond_cat[row * 3 + 2];
    float v = bond_emb[(0 * 16 + c0) * DD + d]
            + bond_emb[(1 * 16 + c1) * DD + d]
            + bond_emb[(2 * 16 + c2) * DD + d];
    float r = rbfb[d];
#pragma unroll
    for (int c = 0; c < 20; ++c) r += e[c] * rbfW[c * DD + d];
    out[(size_t)j * DD + d] = v + r;
}

// -------------------- fused angle-RBF + gather + scatter -------------------
// msg = bond_embed[src] + (RBF(angle)@W + b); agg[dst] += msg
__global__ __launch_bounds__(128) void scatter_ba_kernel(
    const float* __restrict__ bond_embed, // [100000,128]
    const float* __restrict__ angle_float,// [400000]
    const float* __restrict__ rbfW,       // [32,128]
    const float* __restrict__ rbfb,       // [128]
    const int*   __restrict__ src, const int* __restrict__ dst,
    float* __restrict__ agg, int nA) {
    __shared__ float e[32];
    int a = blockIdx.x;
    int d = threadIdx.x;
    if (d < 32) {
        float x = angle_float[a];
        float df = x - 0.1f * (float)d;   // centers 0.0 .. 3.1
        e[d] = expf(-GAMMA * df * df);
    }
    __syncthreads();
    int s = src[a], t = dst[a];
    float r = rbfb[d];
#pragma unroll
    for (int c = 0; c < 32; ++c) r += e[c] * rbfW[c * DD + d];
    float msg = bond_embed[(size_t)s * DD + d] + r;
    atomicAdd(&agg[(size_t)t * DD + d], msg);
}

// -------------------- fused 2-layer MLP + LN + graphnorm + residual --------
// One block = 16 rows. 8 waves: layer1 hidden 16x256 (2 N-tiles/wave) -> LDS,
// layer2 out 16x128 (1 N-tile/wave) -> LDS, then LN/graphnorm/relu/residual.
__global__ __launch_bounds__(256) void mlp_block_kernel(
    const float* __restrict__ agg,    // [N,128]
    const float* __restrict__ w1,     // [128,256]
    const float* __restrict__ b1,     // [256]
    const float* __restrict__ w2,     // [256,128]
    const float* __restrict__ b2,     // [128]
    const float* __restrict__ lng, const float* __restrict__ lnb,
    const float* __restrict__ resid,  // [N,128]
    const int*   __restrict__ gid,    // [N]
    const float* __restrict__ counts, // [N_GRAPHS]
    float* __restrict__ out,          // [N,128] or [2N,128] if repeat2
    int N, int repeat2) {
    __shared__ float lds_in [16 * DD];   //  8 KB
    __shared__ float lds_hid[16 * HH];   // 16 KB
    __shared__ float lds_out[16 * DD];   //  8 KB

    const int row0  = blockIdx.x * 16;
    const int tid   = threadIdx.x;
    const int wave  = tid >> 5;          // 0..7
    const int lane  = tid & 31;
    const int m     = lane & 15;         // M (A) / N (B) index within tile
    const int khalf = lane >> 4;         // K split: 0 or 1

    // stage A tile (16x128) into LDS
    for (int i = tid; i < 16 * DD; i += 256) {
        lds_in[i] = agg[(size_t)(row0 + (i >> 7)) * DD + (i & 127)];
    }
    __syncthreads();

    // ---- layer 1: hidden[16x256] = relu(A @ w1 + b1); wave owns cols
    // [32w, 32w+16) and [32w+16, 32w+32)
    v8f acc0 = {};
    v8f acc1 = {};
    const int col0 = 32 * wave + m;
#pragma unroll 8
    for (int kk = 0; kk < 32; ++kk) {
        const int k = 4 * kk + 2 * khalf;
        v2f a = { lds_in[m * DD + k], lds_in[m * DD + k + 1] };
        v2f bA = { w1[(size_t)k * HH + col0],      w1[(size_t)(k + 1) * HH + col0] };
        v2f bB = { w1[(size_t)k * HH + col0 + 16], w1[(size_t)(k + 1) * HH + col0 + 16] };
        acc0 = wmma_f32(a, bA, acc0);
        acc1 = wmma_f32(a, bB, acc1);
    }
#pragma unroll
    for (int v = 0; v < 8; ++v) {
        const int r = v + 8 * khalf;
        lds_hid[r * HH + col0]      = fmaxf(acc0[v] + b1[col0], 0.0f);
        lds_hid[r * HH + col0 + 16] = fmaxf(acc1[v] + b1[col0 + 16], 0.0f);
    }
    __syncthreads();

    // ---- layer 2: out[16x128] = hidden @ w2 + b2; wave owns cols [16w,16w+16)
    v8f acc = {};
    const int col = 16 * wave + m;
#pragma unroll 8
    for (int kk = 0; kk < 64; ++kk) {
        const int k = 4 * kk + 2 * khalf;
        v2f a = { lds_hid[m * HH + k], lds_hid[m * HH + k + 1] };
        v2f b = { w2[(size_t)k * DD + col], w2[(size_t)(k + 1) * DD + col] };
        acc = wmma_f32(a, b, acc);
    }
#pragma unroll
    for (int v = 0; v < 8; ++v) {
        const int r = v + 8 * khalf;
        lds_out[r * DD + col] = acc[v] + b2[col];
    }
    __syncthreads();

    // ---- epilogue: LayerNorm + sqrt-graphnorm + relu + residual
    // 16 threads per row, 8 channels per thread.
    const int r   = tid >> 4;
    const int sub = tid & 15;
    float vals[8];
    float s = 0.0f;
#pragma unroll
    for (int j = 0; j < 8; ++j) {
        vals[j] = lds_out[r * DD + sub * 8 + j];
        s += vals[j];
    }
#pragma unroll
    for (int off = 1; off < 16; off <<= 1) s += __shfl_xor(s, off, 16);
    const float mu = s * (1.0f / 128.0f);
    float sq = 0.0f;
#pragma unroll
    for (int j = 0; j < 8; ++j) {
        float dd = vals[j] - mu;
        sq += dd * dd;
    }
#pragma unroll
    for (int off = 1; off < 16; off <<= 1) sq += __shfl_xor(sq, off, 16);
    const float rstd = rsqrtf(sq * (1.0f / 128.0f) + LN_EPS);

    const int grow = row0 + r;
    const float gscale = rsqrtf(counts[gid[grow]]);
#pragma unroll
    for (int j = 0; j < 8; ++j) {
        const int c = sub * 8 + j;
        float h = (vals[j] - mu) * rstd * lng[c] + lnb[c];
        h *= gscale;
        h = fmaxf(h, 0.0f);
        h += resid[(size_t)grow * DD + c];
        if (repeat2) {
            out[(size_t)(2 * grow)     * DD + c] = h;
            out[(size_t)(2 * grow + 1) * DD + c] = h;
        } else {
            out[(size_t)grow * DD + c] = h;
        }
    }
}

// ---------------------------------------------------------------------------
extern "C" void kernel_launch(void* const* d_in, const int* in_sizes, int n_in,
                              void* d_out, int out_size, void* d_ws, size_t ws_size,
                              hipStream_t stream) {
    (void)in_sizes; (void)n_in; (void)out_size; (void)ws_size;

    const float* node_feats  = (const float*)d_in[0];
    const float* edge_feats  = (const float*)d_in[1];
    const float* bond_float  = (const float*)d_in[2];
    const float* angle_float = (const float*)d_in[3];
    const float* bond_emb    = (const float*)d_in[4];
    const float* w1a = (const float*)d_in[5];
    const float* b1a = (const float*)d_in[6];
    const float* w2a = (const float*)d_in[7];
    const float* b2a = (const float*)d_in[8];
    const float* lng_a = (const float*)d_in[9];
    const float* lnb_a = (const float*)d_in[10];
    const float* w1b = (const float*)d_in[11];
    const float* b1b = (const float*)d_in[12];
    const float* w2b = (const float*)d_in[13];
    const float* b2b = (const float*)d_in[14];
    const float* lng_b = (const float*)d_in[15];
    const float* lnb_b = (const float*)d_in[16];
    const float* rbfb_W = (const float*)d_in[17];
    const float* rbfb_b = (const float*)d_in[18];
    const float* rbfa_W = (const float*)d_in[19];
    const float* rbfa_b = (const float*)d_in[20];
    const int* ab_src = (const int*)d_in[21];
    const int* ab_dst = (const int*)d_in[22];
    const int* ab_gid = (const int*)d_in[23];
    const int* ba_src = (const int*)d_in[24];
    const int* ba_dst = (const int*)d_in[25];
    const int* ba_gid = (const int*)d_in[26];
    const int* bond_cat = (const int*)d_in[27];

    // workspace layout
    char* ws = (char*)d_ws;
    float* counts_a   = (float*)(ws);                              // 2048
    float* counts_b   = (float*)(ws + 8192);                       // 2048
    float* agg_a      = (float*)(ws + 16384);                      // 100000*128
    float* agg_b      = (float*)(ws + 16384 + (size_t)N_ATOMS * DD * 4);
    float* bond_embed = (float*)(ws + 16384 + 2 * (size_t)N_ATOMS * DD * 4);

    float* node_out = (float*)d_out;                               // [100000,128]
    float* edge_out = (float*)d_out + (size_t)N_ATOMS * DD;        // [200000,128]

    // zero counts + both aggregation buffers (contiguous prefix of ws)
    hipMemsetAsync(d_ws, 0, 16384 + 2 * (size_t)N_ATOMS * DD * 4, stream);

    // per-graph node counts
    count_kernel<<<(N_ATOMS + 255) / 256, 256, 0, stream>>>(ab_gid, counts_a, N_ATOMS);
    count_kernel<<<(N_BONDS + 255) / 256, 256, 0, stream>>>(ba_gid, counts_b, N_BONDS);

    // atom-bond message scatter
    {
        long long total = (long long)N_BONDS_DIR * 32;
        int blocks = (int)((total + 255) / 256);
        scatter_ab_kernel<<<blocks, 256, 0, stream>>>(
            node_feats, edge_feats, ab_src, ab_dst, agg_a, N_BONDS_DIR);
    }

    // bond embedding (cat embeds + bond-length RBF), even directed rows
    bond_embed_kernel<<<N_BONDS, 128, 0, stream>>>(
        bond_emb, bond_cat, bond_float, rbfb_W, rbfb_b, bond_embed, N_BONDS);

    // fused angle-RBF + gather + scatter into bond aggregation
    scatter_ba_kernel<<<N_ANGLES, 128, 0, stream>>>(
        bond_embed, angle_float, rbfa_W, rbfa_b, ba_src, ba_dst, agg_b, N_ANGLES);

    // atom-bond MLP block -> node_out
    mlp_block_kernel<<<N_ATOMS / 16, 256, 0, stream>>>(
        agg_a, w1a, b1a, w2a, b2a, lng_a, lnb_a,
        node_feats, ab_gid, counts_a, node_out, N_ATOMS, 0);

    // bond-angle MLP block -> edge_out (rows repeated x2)
    mlp_block_kernel<<<N_BONDS / 16, 256, 0, stream>>>(
        agg_b, w1b, b1b, w2b, b2b, lng_b, lnb_b,
        bond_embed, ba_gid, counts_b, edge_out, N_BONDS, 1);
}